// CtAslLoss1111_21869973471428
// MI455X (gfx1250) — compile-verified
//
#include <hip/hip_runtime.h>
#include <math.h>

// CDNA5 / gfx1250. wave32. One workgroup per (i,j) pair; GEMM G = E(576x768) @ B^T(768x16)
// via V_WMMA_F32_16X16X4_F32, B-matrix + G + all reductions in LDS.

typedef __attribute__((ext_vector_type(2))) float v2f;
typedef __attribute__((ext_vector_type(8))) float v8f;

#define SB_STRIDE 770   // 768 + 2 pad: spreads LDS banks for B-fragment reads

__global__ __launch_bounds__(256) void ct_kernel(const float* __restrict__ vit,
                                                 const float* __restrict__ bert,
                                                 const int* __restrict__ labels,
                                                 float* __restrict__ ws)
{
    __shared__ __align__(16) float sB[16 * SB_STRIDE];  // 49280 B: rows 0-12 = Lm_j, row 13 = o, 14/15 = 0
    __shared__ __align__(16) float sG[576 * 16];        // 36864 B: G = E @ B^T (col 13 = s)
    __shared__ float sRowE[576];                        // ||E_i||  (first holds squared sums)
    __shared__ float sR[576];                           // r_i = sum_j w_ij beta_j
    __shared__ float sTheta[576];                       // top-32 indicator of s
    __shared__ float sRowL[16];                         // ||Lm_r||
    __shared__ float sBeta[16];                         // softmax(y)
    __shared__ float sYnorm[16];                        // y / sum(y)
    __shared__ float sC[16];                            // c_j = sum_i theta_i w_ij
    __shared__ float sRed[256];
    __shared__ float sFrobL;
    __shared__ float sFrobInv;                          // 1 / (||E||_F * ||Lm||_F)

    const int b   = blockIdx.x;
    const int i   = b / 13;
    const int j   = b - i * 13;
    const int tid = threadIdx.x;

    // ---- beta = softmax(y_i), ynorm = y_i / sum(y_i) ----
    if (tid == 0) {
        float yv[13]; float maxv = -1e30f, ysum = 0.f;
        for (int r = 0; r < 13; ++r) {
            yv[r] = (float)labels[i * 13 + r];
            maxv = fmaxf(maxv, yv[r]);
            ysum += yv[r];
        }
        float esum = 0.f;
        for (int r = 0; r < 13; ++r) { float e = expf(yv[r] - maxv); sBeta[r] = e; esum += e; }
        for (int r = 0; r < 13; ++r) { sBeta[r] /= esum; sYnorm[r] = yv[r] / ysum; }
    }

    // ---- stage Lm_j (13x768) into sB rows 0..12 ----
    for (int idx = tid; idx < 13 * 768; idx += 256) {
        int r = idx / 768; int k = idx - r * 768;
        sB[r * SB_STRIDE + k] = bert[(size_t)((j * 13 + r) * 32) * 768 + k];
    }
    __syncthreads();

    // ---- row 13 = o = Lm^T @ ynorm ; rows 14,15 = 0 ----
    for (int k = tid; k < 768; k += 256) {
        float o = 0.f;
        for (int r = 0; r < 13; ++r) o = fmaf(sYnorm[r], sB[r * SB_STRIDE + k], o);
        sB[13 * SB_STRIDE + k] = o;
        sB[14 * SB_STRIDE + k] = 0.f;
        sB[15 * SB_STRIDE + k] = 0.f;
    }

    // ---- ||Lm_r|| and ||Lm||_F ----
    if (tid < 208) {
        int r = tid >> 4, t16 = tid & 15;
        float p = 0.f;
        for (int k = t16; k < 768; k += 16) { float v = sB[r * SB_STRIDE + k]; p = fmaf(v, v, p); }
        sRed[tid] = p;
    }
    __syncthreads();
    if (tid < 13) {
        float s = 0.f;
        for (int t = 0; t < 16; ++t) s += sRed[tid * 16 + t];
        sRowL[tid] = sqrtf(s);
    }
    __syncthreads();
    if (tid == 0) {
        float fr = 0.f;
        for (int r = 0; r < 13; ++r) fr += sRowL[r] * sRowL[r];
        sFrobL = sqrtf(fr);
    }
    __syncthreads();

    // ---- GEMM: G(576x16) = E(576x768) @ B^T via V_WMMA_F32_16X16X4_F32 ----
    // Fused: per-row squared sums of E (lanes m and m+16 cover complementary column sets).
    const int wave = tid >> 5;
    const int lane = tid & 31;
    const int m    = lane & 15;     // A row within tile / D column index n
    const int half = lane >> 4;

    const size_t ebase = (size_t)((i * 13 + j) * 577 + 1) * 768;  // E = vit[i,j,1:,:]
    for (int t = wave; t < 36; t += 8) {
        const float* Erow = vit + ebase + (size_t)(t * 16 + m) * 768 + 2 * half;
        const float* Brow = &sB[m * SB_STRIDE + 2 * half];
        v8f c = {0.f, 0.f, 0.f, 0.f, 0.f, 0.f, 0.f, 0.f};
        float rsq = 0.f;
#pragma unroll 4
        for (int kk = 0; kk < 768; kk += 4) {
            float2 av = *(const float2*)(Erow + kk);   // A[m][kk+2*half .. +1]
            float2 bv = *(const float2*)(Brow + kk);   // B[kk+2*half .. +1][n=m]
            v2f a = {av.x, av.y};
            v2f bb = {bv.x, bv.y};
            c = __builtin_amdgcn_wmma_f32_16x16x4_f32(
                    /*neg_a=*/false, a, /*neg_b=*/false, bb,
                    /*c_mod=*/(short)0, c, /*reuse_a=*/false, /*reuse_b=*/false);
            rsq = fmaf(av.x, av.x, rsq);
            rsq = fmaf(av.y, av.y, rsq);
        }
        rsq += __shfl_xor(rsq, 16, 32);               // combine the two column halves
        if (half == 0) sRowE[t * 16 + m] = rsq;       // squared for now
#pragma unroll
        for (int v = 0; v < 8; ++v)
            sG[(t * 16 + v + 8 * half) * 16 + m] = c[v];
    }
    __syncthreads();

    // ---- ||E||_F and per-row ||E_i|| ----
    {
        float psum = 0.f;
        for (int ii = tid; ii < 576; ii += 256) psum += sRowE[ii];
        sRed[tid] = psum;
        __syncthreads();
        for (int s = 128; s > 0; s >>= 1) { if (tid < s) sRed[tid] += sRed[tid + s]; __syncthreads(); }
        if (tid == 0) sFrobInv = 1.f / (sqrtf(sRed[0]) * sFrobL);
        __syncthreads();
        for (int ii = tid; ii < 576; ii += 256) sRowE[ii] = sqrtf(sRowE[ii]);
        __syncthreads();
    }

    // ---- theta (exact stable top-32 rank on s = G[:,13]) and r_i ----
    for (int ii = tid; ii < 576; ii += 256) {
        float si = sG[ii * 16 + 13];
        int rank = 0;
        for (int q = 0; q < 576; ++q) {
            float sq = sG[q * 16 + 13];
            rank += (sq > si) ? 1 : ((sq == si && q < ii) ? 1 : 0);
        }
        float th = (rank < 32) ? 1.f : 0.f;
        sTheta[ii] = th;
        float invE = 1.f / sRowE[ii];
        float r = 0.f;
        for (int jj = 0; jj < 13; ++jj) {
            float w = expf(-sG[ii * 16 + jj] * invE / sRowL[jj]);
            r = fmaf(sBeta[jj], w, r);
        }
        sR[ii] = r;
    }
    __syncthreads();

    // ---- c_j = sum_i theta_i w_ij  (13 fixed threads, fixed order -> deterministic) ----
    if (tid < 13) {
        const int jj = tid;
        const float invL = 1.f / sRowL[jj];
        float csum = 0.f;
        for (int ii = 0; ii < 576; ++ii) {
            if (sTheta[ii] != 0.f)
                csum += expf(-sG[ii * 16 + jj] / sRowE[ii] * invL);
        }
        sC[jj] = csum;
    }
    __syncthreads();

    // ---- ct = sum_{i,j} theta_i beta_j w_ij cost_ij (1/r_i + 1/c_j) ----
    float part = 0.f;
    for (int ii = tid; ii < 576; ii += 256) {
        if (sTheta[ii] != 0.f) {
            float invE = 1.f / sRowE[ii];
            float invR = 1.f / sR[ii];
            for (int jj = 0; jj < 13; ++jj) {
                float g = sG[ii * 16 + jj];
                float w = expf(-g * invE / sRowL[jj]);
                float cost = 1.f - g * sFrobInv;
                part = fmaf(sBeta[jj] * w * cost, invR + 1.f / sC[jj], part);
            }
        }
    }
    sRed[tid] = part;
    __syncthreads();
    for (int s = 128; s > 0; s >>= 1) { if (tid < s) sRed[tid] += sRed[tid + s]; __syncthreads(); }
    if (tid == 0) ws[b] = sRed[0];
}

// ---- ASL term: 13 blocks (one per sample i), 13 waves each (one per j) ----
__global__ __launch_bounds__(416) void asl_kernel(const float* __restrict__ vit,
                                                  const float* __restrict__ bert,
                                                  const int* __restrict__ labels,
                                                  float* __restrict__ ws)
{
    const int i    = blockIdx.x;
    const int tid  = threadIdx.x;
    const int wj   = tid >> 5;   // 0..12
    const int lane = tid & 31;

    __shared__ float sT[13];

    const float* x    = vit + (size_t)((i * 13 + 12) * 577) * 768;       // Vi[12, 0]
    const float* Lrow = bert + (size_t)((12 * 13 + wj) * 32) * 768;       // bert_cls[12, wj]

    float d = 0.f;
    for (int k = lane; k < 768; k += 32) d = fmaf(Lrow[k], x[k], d);
#pragma unroll
    for (int off = 16; off > 0; off >>= 1) d += __shfl_down(d, off, 32);

    if (lane == 0) {
        float p  = 1.f / (1.f + expf(-d));
        float yv = (float)labels[i * 13 + wj];
        float term = (yv == 1.f) ? (1.f - p) * logf(p)
                                 : (p * p) * (p * p) * logf(1.f - p);
        sT[wj] = term;
    }
    __syncthreads();
    if (tid == 0) {
        float s = 0.f;
        for (int jj = 0; jj < 13; ++jj) s += sT[jj];
        ws[169 + i] = s;
    }
}

// ---- final deterministic reduction: mean over 13 samples ----
__global__ __launch_bounds__(256) void reduce_kernel(const float* __restrict__ ws,
                                                     float* __restrict__ out)
{
    __shared__ float sRed[256];
    const int tid = threadIdx.x;
    float v = 0.f;
    for (int k = tid; k < 182; k += 256) v += ws[k];
    sRed[tid] = v;
    __syncthreads();
    for (int s = 128; s > 0; s >>= 1) { if (tid < s) sRed[tid] += sRed[tid + s]; __syncthreads(); }
    if (tid == 0) out[0] = sRed[0] / 13.0f;
}

extern "C" void kernel_launch(void* const* d_in, const int* in_sizes, int n_in,
                              void* d_out, int out_size, void* d_ws, size_t ws_size,
                              hipStream_t stream)
{
    const float* vit    = (const float*)d_in[0];   // (13,13,577,768) f32
    const float* bert   = (const float*)d_in[1];   // (13,13,32,768)  f32
    const int*   labels = (const int*)d_in[2];     // (13,13) i32
    float* ws  = (float*)d_ws;                     // 182 floats used
    float* out = (float*)d_out;                    // 1 float

    ct_kernel<<<169, 256, 0, stream>>>(vit, bert, labels, ws);
    asl_kernel<<<13, 416, 0, stream>>>(vit, bert, labels, ws);
    reduce_kernel<<<1, 256, 0, stream>>>(ws, out);
}